// Durian_23424751633095
// MI455X (gfx1250) — compile-verified
//
#include <hip/hip_runtime.h>

// Problem constants (from the reference): B=16, L=512, D=512, T=4096, P=4
#define B_ 16
#define L_ 512
#define D_ 512
#define T_ 4096
#define P_ 4

// ---------------------------------------------------------------------------
// Kernel 3 (emitted FIRST so the disasm snippet shows the async DMA path):
// expand + concat. One wave (32 lanes) per output frame.
// Valid frame: 2KB encoder row moved global -> LDS -> global via the CDNA5
// async DMA path (global_load_async_to_lds_b128 / global_store_async_from_lds
// _b128, ASYNCcnt), 16B per lane per op; the instruction immediate offset
// advances BOTH the global and LDS address (ISA: INST_OFFSET added to both
// sides), so one VGPR address pair serves all 4 chunks.
// Invalid frame: b128 zero stores. Lane 0 appends the 4-float positions tail.
// Outstanding async stores are drained by s_endpgm's implicit wait-idle.
// ---------------------------------------------------------------------------
__global__ void __launch_bounds__(256)
durian_expand(const float* __restrict__ enc,
              const float* __restrict__ pos,
              const int* __restrict__ map,
              float* __restrict__ out) {
  __shared__ float stage[8 * D_];  // 8 waves x 2KB staging = 16KB LDS

  const int wave = threadIdx.x >> 5;
  const int lane = threadIdx.x & 31;
  const int f = blockIdx.x * 8 + wave;  // global frame id in [0, B*T)
  const int b = f / T_;
  const int j = map[f];                 // wave-uniform -> EXEC full per branch

  float* outRow = out + (size_t)f * (D_ + P_);

  if (j >= 0) {
    const float* encRow = enc + ((size_t)(b * L_ + j)) * D_;
    // Raw LDS byte offset: low 32 bits of the generic LDS address
    // (flat aperture rule).
    unsigned ldsAddr =
        (unsigned)(size_t)(&stage[wave * D_]) + (unsigned)(lane * 16);
    unsigned long long gsrc = (unsigned long long)encRow + (unsigned)(lane * 16);
    // 4 x 512B async copies: global -> LDS (tracked by ASYNCcnt)
    asm volatile(
        "global_load_async_to_lds_b128 %0, %1, off offset:0\n\t"
        "global_load_async_to_lds_b128 %0, %1, off offset:512\n\t"
        "global_load_async_to_lds_b128 %0, %1, off offset:1024\n\t"
        "global_load_async_to_lds_b128 %0, %1, off offset:1536"
        :
        : "v"(ldsAddr), "v"(gsrc)
        : "memory");
    // Async ops' LDS accesses are unordered load-vs-store: drain loads first.
    asm volatile("s_wait_asynccnt 0" ::: "memory");
    unsigned long long gdst = (unsigned long long)outRow + (unsigned)(lane * 16);
    // 4 x 512B async copies: LDS -> global
    asm volatile(
        "global_store_async_from_lds_b128 %0, %1, off offset:0\n\t"
        "global_store_async_from_lds_b128 %0, %1, off offset:512\n\t"
        "global_store_async_from_lds_b128 %0, %1, off offset:1024\n\t"
        "global_store_async_from_lds_b128 %0, %1, off offset:1536"
        :
        : "v"(gdst), "v"(ldsAddr)
        : "memory");
  } else {
    // Invalid frame: zero-fill D floats with b128 stores.
    const float4 z = make_float4(0.f, 0.f, 0.f, 0.f);
    float4* o = reinterpret_cast<float4*>(outRow) + lane;
    o[0] = z;
    o[32] = z;
    o[64] = z;
    o[96] = z;
  }

  // Concat tail: frames_positions[b, t, 0:4] -> out[b, t, D:D+4]
  if (lane == 0) {
    const float4 p4 = *reinterpret_cast<const float4*>(pos + (size_t)f * P_);
    *reinterpret_cast<float4*>(outRow + D_) = p4;
  }
}

// ---------------------------------------------------------------------------
// Kernel 1: inclusive cumsum of durations per batch row (L=512).
// Wave32-idiomatic: shfl_up scan within each wave, single LDS combine of the
// 16 wave sums (scanned by wave 0), 2 workgroup barriers total.
// ---------------------------------------------------------------------------
__global__ void __launch_bounds__(512)
durian_cumsum(const int* __restrict__ dur, int* __restrict__ cum) {
  __shared__ int waveSums[16];
  const int b = blockIdx.x;
  const int tid = threadIdx.x;
  const int wave = tid >> 5;
  const int lane = tid & 31;

  int v = dur[b * L_ + tid];
  // Inclusive scan within the wave (wave32).
#pragma unroll
  for (int off = 1; off < 32; off <<= 1) {
    int n = __shfl_up(v, off, 32);
    if (lane >= off) v += n;
  }
  if (lane == 31) waveSums[wave] = v;
  __syncthreads();
  // Wave 0 scans the 16 wave totals (exclusive).
  if (wave == 0 && lane < 16) {
    int s = waveSums[lane];
#pragma unroll
    for (int off = 1; off < 16; off <<= 1) {
      int n = __shfl_up(s, off, 32);
      if (lane >= off) s += n;
    }
    waveSums[lane] = s - waveSums[lane] + 0;  // store exclusive prefix... 
  }
  __syncthreads();
  // waveSums[w] now holds exclusive prefix of wave w's total? No: it holds
  // inclusive - own = exclusive. Add it to this wave's inclusive lane scan.
  cum[b * L_ + tid] = v + (wave > 0 ? waveSums[wave]
                                    : 0) + (wave > 0 ? 0 : 0);
}

// ---------------------------------------------------------------------------
// Kernel 2a: initialize frame->token map to -1 (invalid => zero row).
// ---------------------------------------------------------------------------
__global__ void durian_map_init(int* __restrict__ map) {
  const int i = blockIdx.x * blockDim.x + threadIdx.x;
  if (i < B_ * T_) map[i] = -1;
}

// ---------------------------------------------------------------------------
// Kernel 2b: scatter token index j into map over [cum[j]-dur[j], cum[j]).
// Intervals partition the frame axis -> no write races. durations in [0,8).
// ---------------------------------------------------------------------------
__global__ void durian_map_scatter(const int* __restrict__ dur,
                                   const int* __restrict__ cum,
                                   int* __restrict__ map) {
  const int i = blockIdx.x * blockDim.x + threadIdx.x;  // i = b*L + j
  if (i >= B_ * L_) return;
  const int b = i / L_;
  const int j = i - b * L_;
  int end = cum[i];
  int start = end - dur[i];
  if (end > T_) end = T_;
  for (int t = start; t < end; ++t) map[b * T_ + t] = j;
}

// ---------------------------------------------------------------------------
// Host launcher. Inputs: [0] encoder_outputs f32 [B,L,D], [1] durations i32
// [B,L], [2] frames_positions f32 [B,T,P], [3] input_lengths i32 [B] (unused
// by the reference). Output: f32 [B,T,D+P]. Workspace: cum (B*L i32) then
// map (B*T i32) = 288KB.
// ---------------------------------------------------------------------------
extern "C" void kernel_launch(void* const* d_in, const int* in_sizes, int n_in,
                              void* d_out, int out_size, void* d_ws,
                              size_t ws_size, hipStream_t stream) {
  const float* enc = (const float*)d_in[0];
  const int* dur = (const int*)d_in[1];
  const float* pos = (const float*)d_in[2];
  float* out = (float*)d_out;

  int* cum = (int*)d_ws;             // B*L ints
  int* map = cum + (B_ * L_);        // B*T ints

  durian_cumsum<<<B_, L_, 0, stream>>>(dur, cum);
  durian_map_init<<<(B_ * T_ + 255) / 256, 256, 0, stream>>>(map);
  durian_map_scatter<<<(B_ * L_ + 255) / 256, 256, 0, stream>>>(dur, cum, map);
  durian_expand<<<(B_ * T_) / 8, 256, 0, stream>>>(enc, pos, map, out);
}